// RNNAttention_52261162058016
// MI455X (gfx1250) — compile-verified
//
#include <hip/hip_runtime.h>
#include <hip/hip_bf16.h>

// ---------------- problem dims ----------------
#define UU   512   // LSTM units
#define DIN  256   // input feature dim
#define DATT 512   // attended feature dim
#define BB   32
#define TT   256
#define MMM  512

typedef __attribute__((ext_vector_type(2))) float v2f;
typedef __attribute__((ext_vector_type(8))) float v8f;

// ---------------- fast math helpers ----------------
__device__ __forceinline__ float fast_tanh(float x) {
#if __has_builtin(__builtin_amdgcn_tanhf)
  return __builtin_amdgcn_tanhf(x);   // v_tanh_f32 on gfx1250
#else
  return tanhf(x);
#endif
}
__device__ __forceinline__ float fast_sigm(float x) {
  return 1.0f / (1.0f + __expf(-x));
}

// =====================================================================
// Kernel 1: fp32 GEMM via V_WMMA_F32_16X16X4_F32, 64x16 tile per wave.
// C[M,N] = A[M,K] * B[K,N] + bias[N].
// Each wave owns 4 vertically-stacked 16x16 tiles: the strided B fragment
// is loaded once per k-step and reused by 4 WMMAs; A fragments are
// contiguous 64-bit loads.
// A-frag (16x4 f32): lane=(half<<4)|r -> {A[r, k+2*half], A[r, k+2*half+1]}
// B-frag (4x16 f32): lane=(half<<4)|c -> {B[k+2*half, c], B[k+2*half+1, c]}
// C/D: acc[i] = C[mTile*16 + 8*half + i, nTile*16 + (lane&15)]
// =====================================================================
__launch_bounds__(128)
__global__ void gemm_f32_wmma(const float* __restrict__ A,
                              const float* __restrict__ B,
                              const float* __restrict__ bias,
                              float* __restrict__ C,
                              int Mrows, int N, int K) {
  const int lane   = threadIdx.x & 31;
  const int waveId = blockIdx.x * (blockDim.x >> 5) + (threadIdx.x >> 5);
  const int tilesN = N >> 4;
  const int tileM  = waveId / tilesN;            // 64-row super-tile index
  const int tileN  = waveId - tileM * tilesN;
  if (tileM * 64 >= Mrows) return;

  const int r  = lane & 15;          // row within A tile / col within B tile
  const int kb = (lane >> 4) << 1;   // 0 or 2 : K sub-offset per lane half

  const float* __restrict__ arow0 = A + (size_t)(tileM * 64 + r) * K;
  const size_t aStride16 = (size_t)16 * K;       // 16 rows down
  const float* __restrict__ bcol  = B + (size_t)(tileN * 16 + r);

  v8f acc0 = {}, acc1 = {}, acc2 = {}, acc3 = {};
#pragma unroll 2
  for (int k = 0; k < K; k += 4) {
    v2f b;
    b.x = bcol[(size_t)(k + kb) * N];
    b.y = bcol[(size_t)(k + kb + 1) * N];
    v2f a0, a1, a2, a3;
    a0.x = arow0[k + kb];                  a0.y = arow0[k + kb + 1];
    a1.x = arow0[aStride16     + k + kb];  a1.y = arow0[aStride16     + k + kb + 1];
    a2.x = arow0[aStride16 * 2 + k + kb];  a2.y = arow0[aStride16 * 2 + k + kb + 1];
    a3.x = arow0[aStride16 * 3 + k + kb];  a3.y = arow0[aStride16 * 3 + k + kb + 1];
    acc0 = __builtin_amdgcn_wmma_f32_16x16x4_f32(false, a0, false, b, (short)0, acc0, false, false);
    acc1 = __builtin_amdgcn_wmma_f32_16x16x4_f32(false, a1, false, b, (short)0, acc1, false, false);
    acc2 = __builtin_amdgcn_wmma_f32_16x16x4_f32(false, a2, false, b, (short)0, acc2, false, false);
    acc3 = __builtin_amdgcn_wmma_f32_16x16x4_f32(false, a3, false, b, (short)0, acc3, false, false);
  }

  const int nOut  = tileN * 16 + (lane & 15);
  const int mBase = tileM * 64 + ((lane >> 4) << 3);   // +0 or +8 within each 16-tile
  const float bv  = bias[nOut];
  float* __restrict__ c0 = C + (size_t)mBase * N + nOut;
#pragma unroll
  for (int i = 0; i < 8; ++i) c0[(size_t)i * N]        = acc0[i] + bv;
#pragma unroll
  for (int i = 0; i < 8; ++i) c0[(size_t)(16 + i) * N] = acc1[i] + bv;
#pragma unroll
  for (int i = 0; i < 8; ++i) c0[(size_t)(32 + i) * N] = acc2[i] + bv;
#pragma unroll
  for (int i = 0; i < 8; ++i) c0[(size_t)(48 + i) * N] = acc3[i] + bv;
}

// =====================================================================
// Kernel 2: persistent per-batch LSTM + Bahdanau attention scan.
// grid = 32 (one workgroup per batch), block = 512 threads = 16 waves.
// Recurrence is independent per batch -> only workgroup barriers needed;
// h lives in LDS, c in registers. R/Wq/keys/attended stay L2-resident.
// =====================================================================
__launch_bounds__(512, 1)
__global__ void lstm_attn_scan(const float* __restrict__ xz,       // [B,T,4U]
                               const float* __restrict__ R,        // [U,4U]
                               const float* __restrict__ keys,     // [B,M,U]
                               const float* __restrict__ attended, // [B,M,DATT]
                               const float* __restrict__ Wq,       // [U,U]
                               const float* __restrict__ bq,       // [U]
                               const float* __restrict__ Wa,       // [U]
                               float* __restrict__ seq,            // [B,T,U+DATT]
                               float* __restrict__ hf,             // [B,U]
                               float* __restrict__ cf) {           // [B,U]
  __shared__ float sh_h[UU];
  __shared__ float sh_q[UU];
  __shared__ float sh_w[MMM];
  __shared__ float sh_wa[UU];
  __shared__ float sh_red[16];

  const int b    = blockIdx.x;
  const int tid  = threadIdx.x;
  const int lane = tid & 31;
  const int wave = tid >> 5;

  sh_h[tid]  = 0.0f;
  sh_wa[tid] = Wa[tid];
  float c_state = 0.0f;
  const float bq_t = bq[tid];
  __syncthreads();

  const float* __restrict__ keysB = keys     + (size_t)b * MMM * UU;
  const float* __restrict__ attB  = attended + (size_t)b * MMM * DATT;

  for (int t = 0; t < TT; ++t) {
    // ---- z = xz[b,t] + h @ R ; gate order i,f,g,o ----
    const float* __restrict__ xzt = xz + ((size_t)b * TT + t) * (4 * UU);
    float zi = xzt[tid], zf = xzt[UU + tid], zg = xzt[2 * UU + tid], zo = xzt[3 * UU + tid];
#pragma unroll 4
    for (int u = 0; u < UU; ++u) {
      const float hv = sh_h[u];                       // LDS broadcast
      const float* __restrict__ Rr = R + (size_t)u * (4 * UU);
      zi = fmaf(hv, Rr[tid],          zi);
      zf = fmaf(hv, Rr[UU + tid],     zf);
      zg = fmaf(hv, Rr[2 * UU + tid], zg);
      zo = fmaf(hv, Rr[3 * UU + tid], zo);
    }
    const float ig = fast_sigm(zi);
    const float fg = fast_sigm(zf);
    const float gg = fast_tanh(zg);
    const float og = fast_sigm(zo);
    c_state = fg * c_state + ig * gg;
    const float h_new = og * fast_tanh(c_state);

    __syncthreads();               // everyone done reading old h / old sh_w
    sh_h[tid] = h_new;
    __syncthreads();

    // ---- q = h_new @ Wq + bq ----
    float q = bq_t;
#pragma unroll 4
    for (int u = 0; u < UU; ++u)
      q = fmaf(sh_h[u], Wq[(size_t)u * UU + tid], q);
    sh_q[tid] = q;
    __syncthreads();

    // ---- scores[m] = tanh(keys[b,m,:] + q) . Wa  (wave-cooperative, coalesced) ----
#pragma unroll 2
    for (int mi = 0; mi < 32; ++mi) {                 // 32 rows per wave, 16 waves
      const int m = wave * 32 + mi;
      const float* __restrict__ kr = keysB + (size_t)m * UU;
      float p = 0.0f;
#pragma unroll 4
      for (int u = lane; u < UU; u += 32)
        p = fmaf(fast_tanh(kr[u] + sh_q[u]), sh_wa[u], p);
#pragma unroll
      for (int off = 16; off; off >>= 1) p += __shfl_xor(p, off, 32);
      if (lane == 0) sh_w[m] = p;
    }
    __syncthreads();

    // ---- softmax over M=512 (ba is uniform -> cancels) ----
    const float s = sh_w[tid];
    float r = s;
#pragma unroll
    for (int off = 16; off; off >>= 1) r = fmaxf(r, __shfl_xor(r, off, 32));
    if (lane == 0) sh_red[wave] = r;
    __syncthreads();
    if (wave == 0) {
      float rr = sh_red[lane & 15];
#pragma unroll
      for (int off = 8; off; off >>= 1) rr = fmaxf(rr, __shfl_xor(rr, off, 32));
      if (lane == 0) sh_red[0] = rr;
    }
    __syncthreads();
    const float mx = sh_red[0];
    const float e  = __expf(s - mx);
    __syncthreads();                          // done reading sh_red[0]
    float r2 = e;
#pragma unroll
    for (int off = 16; off; off >>= 1) r2 += __shfl_xor(r2, off, 32);
    if (lane == 0) sh_red[wave] = r2;
    __syncthreads();
    if (wave == 0) {
      float rr = sh_red[lane & 15];
#pragma unroll
      for (int off = 8; off; off >>= 1) rr += __shfl_xor(rr, off, 32);
      if (lane == 0) sh_red[0] = rr;
    }
    __syncthreads();
    sh_w[tid] = e * (1.0f / sh_red[0]);
    __syncthreads();

    // ---- ctx[d] = sum_m w[m] * attended[b,m,d]  (coalesced over d=tid) ----
    float ctx = 0.0f;
#pragma unroll 4
    for (int m = 0; m < MMM; ++m)
      ctx = fmaf(sh_w[m], attB[(size_t)m * DATT + tid], ctx);

    float* __restrict__ outrow = seq + ((size_t)b * TT + t) * (UU + DATT);
    outrow[tid]      = h_new;
    outrow[UU + tid] = ctx;
  }

  hf[(size_t)b * UU + tid] = sh_h[tid];
  cf[(size_t)b * UU + tid] = c_state;
}

// =====================================================================
// launch
// =====================================================================
extern "C" void kernel_launch(void* const* d_in, const int* in_sizes, int n_in,
                              void* d_out, int out_size, void* d_ws, size_t ws_size,
                              hipStream_t stream) {
  (void)in_sizes; (void)n_in; (void)out_size; (void)ws_size;

  const float* inputs   = (const float*)d_in[0];  // [B,T,DIN]
  const float* attended = (const float*)d_in[1];  // [B,M,DATT]
  const float* kernel   = (const float*)d_in[2];  // [DIN,4U]
  const float* rkernel  = (const float*)d_in[3];  // [U,4U]
  const float* bias     = (const float*)d_in[4];  // [4U]
  const float* Wk       = (const float*)d_in[5];  // [DATT,U]
  const float* bk       = (const float*)d_in[6];  // [U]
  const float* Wq       = (const float*)d_in[7];  // [U,U]
  const float* bq       = (const float*)d_in[8];  // [U]
  const float* Wa       = (const float*)d_in[9];  // [U,1]
  // d_in[10] = ba: uniform pre-softmax shift, cancels in softmax.

  float* out = (float*)d_out;
  float* seq = out;                                      // [B,T,U+DATT]
  float* hf  = out + (size_t)BB * TT * (UU + DATT);      // [B,U]
  float* cf  = hf + (size_t)BB * UU;                     // [B,U]

  float* ws_keys = (float*)d_ws;                         // [B,M,U]   32 MB
  float* ws_xz   = ws_keys + (size_t)BB * MMM * UU;      // [B,T,4U]  64 MB

  // keys = attended @ Wk + bk : (B*M=16384) x 512, K=512 ; 64x16 per wave
  {
    const int Mr = BB * MMM, N = UU, K = DATT;
    const int tiles = (Mr / 64) * (N / 16);              // 8192 wave-tiles
    gemm_f32_wmma<<<tiles / 4, 128, 0, stream>>>(attended, Wk, bk, ws_keys, Mr, N, K);
  }
  // xz = inputs @ kernel + bias : (B*T=8192) x 2048, K=256 ; 64x16 per wave
  {
    const int Mr = BB * TT, N = 4 * UU, K = DIN;
    const int tiles = (Mr / 64) * (N / 16);              // 16384 wave-tiles
    gemm_f32_wmma<<<tiles / 4, 128, 0, stream>>>(inputs, kernel, bias, ws_xz, Mr, N, K);
  }
  // sequential scan: one workgroup per batch element
  lstm_attn_scan<<<BB, 512, 0, stream>>>(ws_xz, rkernel, ws_keys, attended,
                                         Wq, bq, Wa, seq, hf, cf);
}